// Encoder_9552007266827
// MI455X (gfx1250) — compile-verified
//
#include <hip/hip_runtime.h>
#include <math.h>

// ---------------- problem constants ----------------
#define T_LEN   36
#define NPOLE   40
#define K4      160            // 4*NPOLE
#define PDIM    4096
#define NITER   100
#define LAM     0.1f
#define TILE_P  64

// LDS strides (floats) chosen for bank-conflict-free WMMA fragment access
#define A_STRIDE  162          // 160x160 A matrix, padded (8B-aligned rows, distinct banks)
#define Y_STRIDE  76           // 160xTILE_P y state, padded
#define YT_STRIDE 72           // 36xTILE_P input tile, padded

// LDS layout (floats)
#define OFF_A    0
#define SZ_A     (K4 * A_STRIDE)            // 25920
#define OFF_Y    (OFF_A + SZ_A)             // also Dt staging (needs 5760 <= 12160)
#define SZ_Y     (K4 * Y_STRIDE)            // 12160
#define OFF_YT   (OFF_Y + SZ_Y)
#define SZ_YT    (T_LEN * YT_STRIDE)        // 2592
#define OFF_TTS  (OFF_YT + SZ_YT)
#define OFF_SC   (OFF_TTS + NITER)
#define SMEM_FLOATS (OFF_SC + 2)            // 40774 -> 163096 bytes

typedef __attribute__((ext_vector_type(2))) float v2f;
typedef __attribute__((ext_vector_type(4))) float v4f;
typedef __attribute__((ext_vector_type(8))) float v8f;

// ---------------------------------------------------------------------------
// Phase 0: build normalized dictionary (transposed), A = I - DtD/||DtD||_F,
// lambda/L, and FISTA momentum table. One block, runs once, negligible cost.
// ---------------------------------------------------------------------------
__global__ __launch_bounds__(256) void setup_kernel(
    const float* __restrict__ Drr, const float* __restrict__ Dth,
    float* __restrict__ wsA, float* __restrict__ wsDt,
    float* __restrict__ wsDtD, float* __restrict__ wsTts,
    float* __restrict__ wsSc)
{
    __shared__ float Dt_s[K4 * T_LEN];
    __shared__ float red[256];
    const int tid = threadIdx.x;

    if (tid < K4) {
        const int g = tid / NPOLE, j = tid % NPOLE;
        const float rr = Drr[j], th = Dth[j];
        float col[T_LEN];
        float sumsq = 0.f;
        float prt = 1.f;                       // rr^t
        for (int t = 0; t < T_LEN; ++t) {
            const float ft = (float)t;
            const float c = cosf(ft * th), s = sinf(ft * th);
            const float sgn = (t & 1) ? -1.f : 1.f;
            float v;
            if      (g == 0) v = prt * c;
            else if (g == 1) v = sgn * prt * c;
            else if (g == 2) v = prt * s;
            else             v = sgn * prt * s;
            col[t] = v;
            sumsq += v * v;
            prt *= rr;
        }
        const float nrm = (sumsq > 0.f) ? sqrtf(sumsq) : sqrtf((float)T_LEN);
        const float inv = 1.f / nrm;
        for (int t = 0; t < T_LEN; ++t) {
            const float v = col[t] * inv;
            Dt_s[tid * T_LEN + t] = v;         // D^T : [160][36]
            wsDt[tid * T_LEN + t] = v;
        }
    }
    __syncthreads();

    // DtD + local sum of squares for Frobenius norm
    float lsum = 0.f;
    for (int e = tid; e < K4 * K4; e += 256) {
        const int i = e / K4, j = e % K4;
        float acc = 0.f;
        for (int t = 0; t < T_LEN; ++t)
            acc += Dt_s[i * T_LEN + t] * Dt_s[j * T_LEN + t];
        wsDtD[e] = acc;
        lsum += acc * acc;
    }
    red[tid] = lsum;
    __syncthreads();
    for (int s = 128; s > 0; s >>= 1) {
        if (tid < s) red[tid] += red[tid + s];
        __syncthreads();
    }
    const float linv = 1.f / sqrtf(red[0]);

    for (int e = tid; e < K4 * K4; e += 256) {
        const int i = e / K4, j = e % K4;
        wsA[e] = ((i == j) ? 1.f : 0.f) - wsDtD[e] * linv;
    }
    if (tid == 0) {
        wsSc[0] = linv;
        wsSc[1] = LAM * linv;
        double tprev = 1.0;
        for (int k = 0; k < NITER; ++k) {
            const double tnext = (1.0 + sqrt(1.0 + 4.0 * tprev * tprev)) * 0.5;
            wsTts[k] = (float)((tprev - 1.0) / tnext);   // (ts[k]-1)/ts[k+1]
            tprev = tnext;
        }
    }
}

// ---------------------------------------------------------------------------
// Fragment helpers for V_WMMA_F32_16X16X4_F32 (wave32)
//  A 16x4 f32: lane<16 -> row m0+lane, cols {k0,k0+1}; lane>=16 -> cols {k0+2,k0+3}
//  B 4x16 f32: symmetric; C/D: VGPR v -> row m0 + v + 8*(lane>=16), col n0 + (lane&15)
// ---------------------------------------------------------------------------
__device__ __forceinline__ v2f load_afrag(const float* base, int stride, int m0,
                                          int k0, int lq, int hl) {
    return *(const v2f*)&base[(m0 + lq) * stride + k0 + 2 * hl];
}
__device__ __forceinline__ v2f load_bfrag(const float* base, int stride, int k0,
                                          int n0, int lq, int hl) {
    v2f b;
    b.x = base[(k0 + 2 * hl) * stride + n0 + lq];
    b.y = base[(k0 + 2 * hl + 1) * stride + n0 + lq];
    return b;
}

// ---------------------------------------------------------------------------
// Phase 1: fused DtY + 100 FISTA iterations, fully on-chip.
// Grid: B * (P/TILE_P) = 128 blocks of 256 threads (8 waves).
// Wave w: N-tile n0=(w&3)*16, M-tiles [(w>>2)*5 .. +5) of the 160-row output.
// ---------------------------------------------------------------------------
__global__ __launch_bounds__(256, 1) void fista_kernel(
    const float* __restrict__ X, const float* __restrict__ wsA,
    const float* __restrict__ wsDt, const float* __restrict__ wsTts,
    const float* __restrict__ wsSc, float* __restrict__ out)
{
    extern __shared__ float smem[];
    float* A_s   = smem + OFF_A;
    float* y_s   = smem + OFF_Y;     // also Dt staging during setup
    float* yt_s  = smem + OFF_YT;
    float* tts_s = smem + OFF_TTS;
    float* sc_s  = smem + OFF_SC;

    const int tid  = threadIdx.x;
    const int lane = tid & 31;
    const int w    = tid >> 5;
    const int hl   = lane >> 4;          // lane half (0/1)
    const int lq   = lane & 15;
    const int b    = blockIdx.x >> 6;    // P/TILE_P = 64 tiles per batch
    const int p0   = (blockIdx.x & 63) * TILE_P;
    const int n0   = (w & 3) * 16;
    const int mb   = (w >> 2) * 5;       // first M-tile index for this wave

    // ---- cooperative staging ----
    for (int i = tid; i < K4 * K4; i += 256) {      // A with row padding
        const int r = i / K4, c = i % K4;
        A_s[r * A_STRIDE + c] = wsA[i];
    }
    {                                                // D^T (flat) into y region
        const v4f* src = (const v4f*)wsDt;
        v4f* dst = (v4f*)y_s;
        for (int i = tid; i < (K4 * T_LEN) / 4; i += 256) dst[i] = src[i];
    }
    for (int i = tid; i < T_LEN * TILE_P; i += 256) { // input tile [36][64]
        const int t = i >> 6, c = i & 63;
        yt_s[t * YT_STRIDE + c] = X[(b * T_LEN + t) * PDIM + p0 + c];
    }
    if (tid < NITER) tts_s[tid] = wsTts[tid];
    if (tid < 2) sc_s[tid] = wsSc[tid];
    __syncthreads();

    const float linv  = sc_s[0];
    const float lambd = sc_s[1];

    // ---- DtY = (D^T @ Ytile) * linv  via WMMA f32, K = 36 ----
    v8f dty[5];
    for (int i = 0; i < 5; ++i)
        for (int e = 0; e < 8; ++e) dty[i][e] = 0.f;

    for (int k0 = 0; k0 < T_LEN; k0 += 4) {
        const v2f bf = load_bfrag(yt_s, YT_STRIDE, k0, n0, lq, hl);
        for (int i = 0; i < 5; ++i) {
            const v2f af = load_afrag(y_s, T_LEN, (mb + i) * 16, k0, lq, hl);
            dty[i] = __builtin_amdgcn_wmma_f32_16x16x4_f32(
                false, af, false, bf, (short)0, dty[i], false, false);
        }
    }

    // ---- iteration 0: x = shrink(DtY, lambda); y = x (tt[0] == 0) ----
    v8f x_old[5];
    for (int i = 0; i < 5; ++i)
        for (int e = 0; e < 8; ++e) {
            const float z = dty[i][e] * linv;
            dty[i][e] = z;
            const float t = fabsf(z) - lambd;
            x_old[i][e] = (t > 0.f) ? copysignf(t, z) : 0.f;
        }
    __syncthreads();                      // Dt staging in y_s no longer needed
    for (int i = 0; i < 5; ++i) {
        const int m0 = (mb + i) * 16;
        for (int v = 0; v < 8; ++v)
            y_s[(m0 + v + 8 * hl) * Y_STRIDE + n0 + lq] = x_old[i][v];
    }
    __syncthreads();

    // ---- iterations 1..99: z = A@y + DtY; x=shrink(z); y=(1+tt)x - tt*x_old ----
    for (int it = 1; it < NITER; ++it) {
        const float tt = tts_s[it];
        v8f acc[5];
        for (int i = 0; i < 5; ++i) acc[i] = dty[i];   // C starts at DtY

        for (int k0 = 0; k0 < K4; k0 += 4) {
            const v2f bf = load_bfrag(y_s, Y_STRIDE, k0, n0, lq, hl);
            for (int i = 0; i < 5; ++i) {
                const v2f af = load_afrag(A_s, A_STRIDE, (mb + i) * 16, k0, lq, hl);
                acc[i] = __builtin_amdgcn_wmma_f32_16x16x4_f32(
                    false, af, false, bf, (short)0, acc[i], false, false);
            }
        }
        __syncthreads();                  // all reads of y_s complete
        for (int i = 0; i < 5; ++i) {
            const int m0 = (mb + i) * 16;
            for (int v = 0; v < 8; ++v) {
                const float z  = acc[i][v];
                const float t  = fabsf(z) - lambd;
                const float xn = (t > 0.f) ? copysignf(t, z) : 0.f;
                const float yn = xn * (1.f + tt) - x_old[i][v] * tt;
                x_old[i][v] = xn;
                y_s[(m0 + v + 8 * hl) * Y_STRIDE + n0 + lq] = yn;
            }
        }
        __syncthreads();
    }

    // ---- store final x: out[b][row][p] ----
    for (int i = 0; i < 5; ++i) {
        const int m0 = (mb + i) * 16;
        for (int v = 0; v < 8; ++v) {
            const int r = m0 + v + 8 * hl;
            out[((size_t)b * K4 + r) * PDIM + p0 + n0 + lq] = x_old[i][v];
        }
    }
}

// ---------------------------------------------------------------------------
extern "C" void kernel_launch(void* const* d_in, const int* in_sizes, int n_in,
                              void* d_out, int out_size, void* d_ws, size_t ws_size,
                              hipStream_t stream) {
    (void)in_sizes; (void)n_in; (void)out_size; (void)ws_size;
    const float* Drr = (const float*)d_in[0];
    const float* Dth = (const float*)d_in[1];
    const float* X   = (const float*)d_in[2];

    float* ws     = (float*)d_ws;
    float* wsA    = ws;                       // 160*160
    float* wsDt   = ws + 25600;               // 160*36
    float* wsDtD  = ws + 31360;               // 160*160 (scratch)
    float* wsTts  = ws + 56960;               // 100
    float* wsSc   = ws + 57060;               // {1/L, lam/L}

    setup_kernel<<<1, 256, 0, stream>>>(Drr, Dth, wsA, wsDt, wsDtD, wsTts, wsSc);

    const size_t smemBytes = (size_t)SMEM_FLOATS * sizeof(float);  // 163096 B
    const int nblocks = 2 * (PDIM / TILE_P);                        // 128
    fista_kernel<<<nblocks, 256, smemBytes, stream>>>(
        X, wsA, wsDt, wsTts, wsSc, (float*)d_out);
}